// DepthUpdate_35639638622604
// MI455X (gfx1250) — compile-verified
//
#include <hip/hip_runtime.h>
#include <math.h>

// ---------------- sizes ----------------
#define Bn   2
#define Cc   32
#define Dd   32
#define Hh   128
#define Ww   160
#define HW   (Hh*Ww)     // 20480
#define Ee   32
#define NQn  32
#define NHh  4
#define Ss   320         // (128/8)*(160/8)
#define TOK  (Ss*Bn)     // 640 tokens

typedef float v2f __attribute__((ext_vector_type(2)));
typedef float v4f __attribute__((ext_vector_type(4)));
typedef float v8f __attribute__((ext_vector_type(8)));

// ---------------- 1: d-map + entropy mask ----------------
__global__ __launch_bounds__(256) void precompute_kernel(
    const float* __restrict__ depth, const float* __restrict__ prob,
    const float* __restrict__ dip_p, const int* __restrict__ nd_p,
    float* __restrict__ dmap, float* __restrict__ mask) {
  int idx = blockIdx.x * 256 + threadIdx.x;       // Bn*HW
  float dip = dip_p[0];
  int   nd  = nd_p[0];
  float dv   = depth[idx];
  float dmin = dv - 0.5f * (float)nd * dip;
  float dmax = dv + 0.5f * (float)nd * dip;
  float d    = (dv - dmin) / (dmax - dmin + 1e-10f);
  dmap[idx]  = d;
  int n = idx / HW, p = idx % HW;
  float s = 0.0f;
  for (int k = 0; k < Dd; ++k) {
    float pv = prob[(n * Dd + k) * HW + p];
    s += pv * logf(fminf(fmaxf(pv, 1e-9f), 1.0f));
  }
  float entropy = s / (-logf((float)nd));
  mask[idx] = (1.0f - entropy > 0.3f) ? 1.0f : 0.0f;
}

// ---------------- 2: cost = mask * sum_d cv*pv  (the 167MB pass) ----------------
// cv is streamed once -> non-temporal loads so the reused 5MB prob volume
// stays resident in the 192MB L2 instead of being churned by the stream.
__global__ __launch_bounds__(256) void cost_reduce_kernel(
    const float* __restrict__ cv, const float* __restrict__ pv,
    const float* __restrict__ mask, float* __restrict__ cost) {
  int idx = blockIdx.x * 256 + threadIdx.x;       // Bn*Cc*(HW/4)
  int p4 = idx % (HW / 4);
  int c  = (idx / (HW / 4)) % Cc;
  int n  = idx / ((HW / 4) * Cc);
  int p  = p4 * 4;
  const v4f* cvp = (const v4f*)(cv + (size_t)((n * Cc + c) * Dd) * HW + p);
  const v4f* pvp = (const v4f*)(pv + (size_t)(n * Dd) * HW + p);
  v4f acc = {0.f, 0.f, 0.f, 0.f};
  #pragma unroll 4
  for (int d = 0; d < Dd; ++d) {
    v4f a = __builtin_nontemporal_load(&cvp[d * (HW / 4)]);   // single-use stream
    v4f b = pvp[d * (HW / 4)];                                // L2-resident reuse
    acc += a * b;
  }
  v4f m = *(const v4f*)(mask + n * HW + p);
  acc *= m;
  *(v4f*)(cost + (size_t)(n * Cc + c) * HW + p) = acc;
}

// ---------------- 3: patch embed (8x8 stride-8 conv) + pos encoding ----------------
__global__ __launch_bounds__(256) void patch_embed_kernel(
    const float* __restrict__ rf, const float* __restrict__ dmap,
    const float* __restrict__ wpxp, const float* __restrict__ bpxp,
    float* __restrict__ t) {
  __shared__ float partial[256];
  int tokflat = blockIdx.x;              // s*2 + n
  int n = tokflat & 1, s = tokflat >> 1;
  int py = s / 20, px = s % 20;
  int e = threadIdx.x & 31, part = threadIdx.x >> 5;
  float acc = 0.0f;
  for (int i = part * 264; i < (part + 1) * 264; ++i) {   // 33*64 = 2112 / 8 parts
    int c = i >> 6, rem = i & 63;
    int iy = rem >> 3, ix = rem & 7;
    int hh = py * 8 + iy, xx = px * 8 + ix;
    float xv = (c < Cc) ? rf[(n * Cc + c) * HW + hh * Ww + xx]
                        : dmap[n * HW + hh * Ww + xx];
    acc += xv * wpxp[(e * 33 + c) * 64 + rem];
  }
  partial[threadIdx.x] = acc;
  __syncthreads();
  if (threadIdx.x < 32) {
    float sum = 0.0f;
    #pragma unroll
    for (int pg = 0; pg < 8; ++pg) sum += partial[pg * 32 + e];
    sum += bpxp[e];
    int k = e >> 1;
    float den = expf(-(float)k * (logf(10000.0f) / 16.0f));
    float arg = (float)s * den;
    sum += (e & 1) ? cosf(arg) : sinf(arg);
    t[tokflat * Ee + e] = sum;
  }
}

// ---------------- 4: qkv projection ----------------
__global__ __launch_bounds__(256) void qkv_kernel(
    const float* __restrict__ t, const float* __restrict__ wqkv,
    const float* __restrict__ bqkv, float* __restrict__ qkv) {
  int idx = blockIdx.x * 256 + threadIdx.x;   // TOK*96
  int f = idx % 96, tok = idx / 96;
  const float* trow = t + tok * Ee;
  const float* wr   = wqkv + f * Ee;
  float acc = bqkv[f];
  #pragma unroll
  for (int e = 0; e < Ee; ++e) acc += trow[e] * wr[e];
  qkv[idx] = acc;
}

// ---------------- 5: attention (wave per (s,n,h) row, streaming softmax) ----------------
__global__ __launch_bounds__(256) void attn_kernel(
    const float* __restrict__ qkv, float* __restrict__ o) {
  int wid  = (blockIdx.x * 256 + threadIdx.x) >> 5;  // 0..2559
  int lane = threadIdx.x & 31;
  int h = wid % NHh, n = (wid / NHh) % Bn, s = wid / (NHh * Bn);
  const float inv = 0.35355339059327373f;            // 1/sqrt(8)
  float q[8];
  #pragma unroll
  for (int d = 0; d < 8; ++d) q[d] = qkv[(s * Bn + n) * 96 + h * 8 + d] * inv;
  float sc[10];
  #pragma unroll
  for (int it = 0; it < 10; ++it) {
    int tt = it * 32 + lane;
    const float* kp = qkv + (tt * Bn + n) * 96 + 32 + h * 8;
    float a = 0.f;
    #pragma unroll
    for (int d = 0; d < 8; ++d) a += q[d] * kp[d];
    sc[it] = a;
  }
  float m = sc[0];
  #pragma unroll
  for (int it = 1; it < 10; ++it) m = fmaxf(m, sc[it]);
  for (int off = 16; off; off >>= 1) m = fmaxf(m, __shfl_xor(m, off, 32));
  float pr[10], ssum = 0.f;
  #pragma unroll
  for (int it = 0; it < 10; ++it) { pr[it] = expf(sc[it] - m); ssum += pr[it]; }
  for (int off = 16; off; off >>= 1) ssum += __shfl_xor(ssum, off, 32);
  float acc[8] = {0.f, 0.f, 0.f, 0.f, 0.f, 0.f, 0.f, 0.f};
  #pragma unroll
  for (int it = 0; it < 10; ++it) {
    int tt = it * 32 + lane;
    const float* vp = qkv + (tt * Bn + n) * 96 + 64 + h * 8;
    #pragma unroll
    for (int d = 0; d < 8; ++d) acc[d] += pr[it] * vp[d];
  }
  #pragma unroll
  for (int d = 0; d < 8; ++d)
    for (int off = 16; off; off >>= 1) acc[d] += __shfl_xor(acc[d], off, 32);
  if (lane == 0) {
    float rs = 1.0f / ssum;
    #pragma unroll
    for (int d = 0; d < 8; ++d) o[(s * Bn + n) * Ee + h * 8 + d] = acc[d] * rs;
  }
}

// ---------------- 6: out-proj + residual + layernorm (wave per token) ----------------
__global__ __launch_bounds__(256) void proj_ln_kernel(
    const float* __restrict__ o, const float* __restrict__ wo,
    const float* __restrict__ bo, const float* __restrict__ g,
    const float* __restrict__ b, float* __restrict__ t) {
  int tok = (blockIdx.x * 256 + threadIdx.x) >> 5;   // 0..639
  int e = threadIdx.x & 31;
  const float* orow = o + tok * Ee;
  float val = bo[e] + t[tok * Ee + e];
  #pragma unroll
  for (int e2 = 0; e2 < Ee; ++e2) val += orow[e2] * wo[e * Ee + e2];
  float mean = val;
  for (int off = 16; off; off >>= 1) mean += __shfl_xor(mean, off, 32);
  mean *= (1.0f / 32.0f);
  float dv = val - mean;
  float var = dv * dv;
  for (int off = 16; off; off >>= 1) var += __shfl_xor(var, off, 32);
  var *= (1.0f / 32.0f);
  t[tok * Ee + e] = dv * rsqrtf(var + 1e-5f) * g[e] + b[e];
}

// ---------------- 7: feedforward (32->64 relu ->32) + residual + LN ----------------
__global__ __launch_bounds__(256) void ff_kernel(
    const float* __restrict__ w1, const float* __restrict__ b1,
    const float* __restrict__ w2, const float* __restrict__ b2,
    const float* __restrict__ g, const float* __restrict__ bb,
    float* __restrict__ t) {
  __shared__ float ff[4][64];
  int grp = threadIdx.x >> 6, wg = threadIdx.x & 63;
  int tok = blockIdx.x * 4 + grp;
  const float* trow = t + tok * Ee;
  float hv = b1[wg];
  #pragma unroll
  for (int e = 0; e < Ee; ++e) hv += trow[e] * w1[wg * Ee + e];
  ff[grp][wg] = fmaxf(hv, 0.0f);
  __syncthreads();
  if (wg < 32) {
    int e = wg;
    float val = b2[e] + trow[e];
    #pragma unroll
    for (int f = 0; f < 64; ++f) val += ff[grp][f] * w2[e * 64 + f];
    float mean = val;
    for (int off = 16; off; off >>= 1) mean += __shfl_xor(mean, off, 32);
    mean *= (1.0f / 32.0f);
    float dv = val - mean;
    float var = dv * dv;
    for (int off = 16; off; off >>= 1) var += __shfl_xor(var, off, 32);
    var *= (1.0f / 32.0f);
    t[tok * Ee + e] = dv * rsqrtf(var + 1e-5f) * g[e] + bb[e];
  }
}

// ---------------- 8: generic direct 3x3 conv (optional d-map ch, bias, bn_relu) ----------------
__global__ __launch_bounds__(256) void conv33_kernel(
    const float* __restrict__ in32, int Cin0,
    const float* __restrict__ dmap,               // null => Cin = Cin0
    const float* __restrict__ w,
    const float* __restrict__ bias,               // nullable
    const float* __restrict__ bng, const float* __restrict__ bnb,  // nullable
    float* __restrict__ out, int Cout) {
  int idx = blockIdx.x * 256 + threadIdx.x;       // Bn*Cout*HW
  int p  = idx % HW;
  int co = (idx / HW) % Cout;
  int n  = idx / (HW * Cout);
  int hy = p / Ww, hx = p % Ww;
  float acc = bias ? bias[co] : 0.0f;
  int Cin = Cin0 + (dmap ? 1 : 0);
  for (int c = 0; c < Cin; ++c) {
    const float* src = (c < Cin0) ? in32 + (size_t)(n * Cin0 + c) * HW : dmap + n * HW;
    const float* wr  = w + (co * Cin + c) * 9;
    #pragma unroll
    for (int dy = -1; dy <= 1; ++dy) {
      int yy = hy + dy; if ((unsigned)yy >= (unsigned)Hh) continue;
      #pragma unroll
      for (int dx = -1; dx <= 1; ++dx) {
        int xx = hx + dx; if ((unsigned)xx >= (unsigned)Ww) continue;
        acc += src[yy * Ww + xx] * wr[(dy + 1) * 3 + (dx + 1)];
      }
    }
  }
  if (bng) acc = fmaxf(acc * (bng[co] * rsqrtf(1.0f + 1e-5f)) + bnb[co], 0.0f);
  out[idx] = acc;
}

// ---------------- 9: ref_attn = queries x feat  (WMMA f32 16x16x4, M=K=32) ----------------
__global__ __launch_bounds__(256) void refattn_wmma_kernel(
    const float* __restrict__ feat, const float* __restrict__ tbuf,
    float* __restrict__ rattn) {
  int wave = (blockIdx.x * 256 + threadIdx.x) >> 5;   // 0..2559
  int lane = threadIdx.x & 31;
  int n = wave / (HW / 16);
  int p0 = (wave % (HW / 16)) * 16;
  int col = lane & 15;
  int hi  = lane >> 4;                 // K half select (0 -> K{0,1}, 1 -> K{2,3})
  v8f acc0 = {0.f,0.f,0.f,0.f,0.f,0.f,0.f,0.f};
  v8f acc1 = acc0;
  #pragma unroll
  for (int ks = 0; ks < 8; ++ks) {
    int k0 = ks * 4 + hi * 2;
    v2f b;
    b.x = feat[(size_t)(n * Ee + (k0 + 0)) * HW + p0 + col];
    b.y = feat[(size_t)(n * Ee + (k0 + 1)) * HW + p0 + col];
    v2f a0, a1;
    a0.x = tbuf[(col) * (Bn * Ee) + n * Ee + (k0 + 0)];        // queries[q=col][n][k]
    a0.y = tbuf[(col) * (Bn * Ee) + n * Ee + (k0 + 1)];
    a1.x = tbuf[(col + 16) * (Bn * Ee) + n * Ee + (k0 + 0)];
    a1.y = tbuf[(col + 16) * (Bn * Ee) + n * Ee + (k0 + 1)];
    acc0 = __builtin_amdgcn_wmma_f32_16x16x4_f32(false, a0, false, b, (short)0, acc0, false, false);
    acc1 = __builtin_amdgcn_wmma_f32_16x16x4_f32(false, a1, false, b, (short)0, acc1, false, false);
  }
  int rb = hi * 8;
  #pragma unroll
  for (int j = 0; j < 8; ++j) {
    rattn[(size_t)(n * NQn + (rb + j)) * HW + p0 + col]      = acc0[j];
    rattn[(size_t)(n * NQn + (16 + rb + j)) * HW + p0 + col] = acc1[j];
  }
}

// ---------------- 10: u = relu(bn(w_cost x cost)) + ref_attn   (WMMA) ----------------
__global__ __launch_bounds__(256) void costconv_wmma_kernel(
    const float* __restrict__ cost, const float* __restrict__ wc,
    const float* __restrict__ gq, const float* __restrict__ bq,
    const float* __restrict__ rattn, float* __restrict__ u) {
  int wave = (blockIdx.x * 256 + threadIdx.x) >> 5;
  int lane = threadIdx.x & 31;
  int n = wave / (HW / 16);
  int p0 = (wave % (HW / 16)) * 16;
  int col = lane & 15;
  int hi  = lane >> 4;
  v8f acc0 = {0.f,0.f,0.f,0.f,0.f,0.f,0.f,0.f};
  v8f acc1 = acc0;
  #pragma unroll
  for (int ks = 0; ks < 8; ++ks) {
    int k0 = ks * 4 + hi * 2;
    v2f b;
    b.x = cost[(size_t)(n * Cc + (k0 + 0)) * HW + p0 + col];
    b.y = cost[(size_t)(n * Cc + (k0 + 1)) * HW + p0 + col];
    v2f a0, a1;
    a0.x = wc[(col) * Cc + (k0 + 0)];
    a0.y = wc[(col) * Cc + (k0 + 1)];
    a1.x = wc[(col + 16) * Cc + (k0 + 0)];
    a1.y = wc[(col + 16) * Cc + (k0 + 1)];
    acc0 = __builtin_amdgcn_wmma_f32_16x16x4_f32(false, a0, false, b, (short)0, acc0, false, false);
    acc1 = __builtin_amdgcn_wmma_f32_16x16x4_f32(false, a1, false, b, (short)0, acc1, false, false);
  }
  float bns = rsqrtf(1.0f + 1e-5f);
  int rb = hi * 8;
  #pragma unroll
  for (int j = 0; j < 8; ++j) {
    int q0 = rb + j, q1 = 16 + rb + j;
    size_t i0 = (size_t)(n * NQn + q0) * HW + p0 + col;
    size_t i1 = (size_t)(n * NQn + q1) * HW + p0 + col;
    u[i0] = fmaxf(acc0[j] * (gq[q0] * bns) + bq[q0], 0.0f) + rattn[i0];
    u[i1] = fmaxf(acc1[j] * (gq[q1] * bns) + bq[q1], 0.0f) + rattn[i1];
  }
}

// ---------------- 11: final 8->1 conv + depth denorm ----------------
__global__ __launch_bounds__(256) void final_kernel(
    const float* __restrict__ t1, const float* __restrict__ w2,
    const float* __restrict__ dmap, const float* __restrict__ depth,
    const float* __restrict__ dip_p, const int* __restrict__ nd_p,
    float* __restrict__ out) {
  int idx = blockIdx.x * 256 + threadIdx.x;      // Bn*HW
  int p = idx % HW, n = idx / HW;
  int hy = p / Ww, hx = p % Ww;
  float acc = 0.0f;
  for (int c = 0; c < 8; ++c) {
    const float* src = t1 + (size_t)(n * 8 + c) * HW;
    const float* wr  = w2 + c * 9;
    #pragma unroll
    for (int dy = -1; dy <= 1; ++dy) {
      int yy = hy + dy; if ((unsigned)yy >= (unsigned)Hh) continue;
      #pragma unroll
      for (int dx = -1; dx <= 1; ++dx) {
        int xx = hx + dx; if ((unsigned)xx >= (unsigned)Ww) continue;
        acc += src[yy * Ww + xx] * wr[(dy + 1) * 3 + (dx + 1)];
      }
    }
  }
  float dip = dip_p[0];
  float nd  = (float)nd_p[0];
  float dmin  = depth[idx] - 0.5f * nd * dip;
  float range = nd * dip;
  out[idx] = (dmap[idx] + acc) * range + dmin;
}

// ---------------- host ----------------
extern "C" void kernel_launch(void* const* d_in, const int* in_sizes, int n_in,
                              void* d_out, int out_size, void* d_ws, size_t ws_size,
                              hipStream_t stream) {
  const float* ref_feat = (const float*)d_in[0];
  const float* cost_vol = (const float*)d_in[1];
  const float* prob_vol = (const float*)d_in[2];
  const float* depth    = (const float*)d_in[3];
  const float* dip      = (const float*)d_in[4];
  const float* w_cost   = (const float*)d_in[5];
  const float* g_cost   = (const float*)d_in[6];
  const float* b_cost   = (const float*)d_in[7];
  const float* w0       = (const float*)d_in[8];
  const float* g0       = (const float*)d_in[9];
  const float* b0       = (const float*)d_in[10];
  const float* w1c      = (const float*)d_in[11];
  const float* w2c      = (const float*)d_in[12];
  const float* w3x3     = (const float*)d_in[13];
  const float* b3x3     = (const float*)d_in[14];
  const float* wpxp     = (const float*)d_in[15];
  const float* bpxp     = (const float*)d_in[16];
  const float* t_wqkv   = (const float*)d_in[17];
  const float* t_bqkv   = (const float*)d_in[18];
  const float* t_wo     = (const float*)d_in[19];
  const float* t_bo     = (const float*)d_in[20];
  const float* t_ln1g   = (const float*)d_in[21];
  const float* t_ln1b   = (const float*)d_in[22];
  const float* t_w1     = (const float*)d_in[23];
  const float* t_b1     = (const float*)d_in[24];
  const float* t_w2     = (const float*)d_in[25];
  const float* t_b2     = (const float*)d_in[26];
  const float* t_ln2g   = (const float*)d_in[27];
  const float* t_ln2b   = (const float*)d_in[28];
  const int*   ndepth   = (const int*)d_in[29];
  float* out = (float*)d_out;
  (void)in_sizes; (void)n_in; (void)out_size; (void)ws_size;

  // workspace layout (floats)
  float* W     = (float*)d_ws;
  float* dmap  = W;                         // 40960
  float* mask  = dmap  + (size_t)Bn * HW;   // 40960
  float* cost  = mask  + (size_t)Bn * HW;   // 1310720
  float* tbuf  = cost  + (size_t)Bn * Cc * HW;   // 20480
  float* qkvb  = tbuf  + (size_t)TOK * Ee;       // 61440
  float* obuf  = qkvb  + (size_t)TOK * 96;       // 20480
  float* feat  = obuf  + (size_t)TOK * Ee;       // 1310720
  float* rattn = feat  + (size_t)Bn * Ee * HW;   // 1310720
  float* ubuf  = rattn + (size_t)Bn * NQn * HW;  // 1310720
  float* t0 = cost;   // reuse: cost dead after costconv_wmma
  float* t1 = feat;   // reuse: feat dead after refattn_wmma

  // cost path
  precompute_kernel<<<(Bn*HW)/256, 256, 0, stream>>>(depth, prob_vol, dip, ndepth, dmap, mask);
  cost_reduce_kernel<<<(Bn*Cc*HW/4)/256, 256, 0, stream>>>(cost_vol, prob_vol, mask, cost);

  // transformer path
  patch_embed_kernel<<<TOK, 256, 0, stream>>>(ref_feat, dmap, wpxp, bpxp, tbuf);
  for (int l = 0; l < 2; ++l) {
    qkv_kernel<<<(TOK*96)/256, 256, 0, stream>>>(tbuf, t_wqkv + l*96*Ee, t_bqkv + l*96, qkvb);
    attn_kernel<<<(Ss*Bn*NHh*32)/256, 256, 0, stream>>>(qkvb, obuf);
    proj_ln_kernel<<<(TOK*32)/256, 256, 0, stream>>>(obuf, t_wo + l*Ee*Ee, t_bo + l*Ee,
                                                     t_ln1g + l*Ee, t_ln1b + l*Ee, tbuf);
    ff_kernel<<<TOK/4, 256, 0, stream>>>(t_w1 + l*64*Ee, t_b1 + l*64,
                                         t_w2 + l*Ee*64, t_b2 + l*Ee,
                                         t_ln2g + l*Ee, t_ln2b + l*Ee, tbuf);
  }

  // feat conv (33ch in, 32 out, +bias) then WMMA query attention
  conv33_kernel<<<(Bn*Ee*HW)/256, 256, 0, stream>>>(ref_feat, Cc, dmap, w3x3, b3x3,
                                                    nullptr, nullptr, feat, Ee);
  refattn_wmma_kernel<<<(Bn*(HW/16)*32)/256, 256, 0, stream>>>(feat, tbuf, rattn);

  // 1x1 cost conv + bn_relu + add attn (WMMA)
  costconv_wmma_kernel<<<(Bn*(HW/16)*32)/256, 256, 0, stream>>>(cost, w_cost, g_cost, b_cost,
                                                                rattn, ubuf);

  // residual conv stack
  conv33_kernel<<<(Bn*16*HW)/256, 256, 0, stream>>>(ubuf, NQn, dmap, w0, nullptr,
                                                    g0, b0, t0, 16);
  conv33_kernel<<<(Bn*8*HW)/256, 256, 0, stream>>>(t0, 16, nullptr, w1c, nullptr,
                                                   nullptr, nullptr, t1, 8);
  final_kernel<<<(Bn*HW)/256, 256, 0, stream>>>(t1, w2c, dmap, depth, dip, ndepth, out);
}